// CPC_Loss_21715354648691
// MI455X (gfx1250) — compile-verified
//
#include <hip/hip_runtime.h>
#include <stdint.h>

// Problem constants (from reference): B,T,E,S,K
#define Bc 64
#define Tc 512
#define Ec 256
#define Sc 5
#define Kc 128

typedef __attribute__((ext_vector_type(2))) float v2f;
typedef __attribute__((ext_vector_type(8))) float v8f;

// ---- LDS layout (in floats) ----
// negB : 256x128 stored as float4 rows: neg4[(e>>2)*128 + k] = {v[e..e+3]}
//        (matches GLOBAL_LOAD_ASYNC_TO_LDS_B128: one float4 per lane per op)
// aB   : 16 rows x pitch 260 (pad for bank-conflict-free WMMA A fragment loads)
// lg   : 16 rows x pitch 132 logits (col 0 = pos, cols 1..128 = neg)
// red  : 16x8 reduction scratch
// acc2 : block-level {loss_sum, count}
#define NEG_OFF   0
#define A_OFF     (Ec * Kc)                 // 32768
#define A_PITCH   260
#define LG_OFF    (A_OFF + 16 * A_PITCH)    // 36928
#define LG_PITCH  132
#define RED_OFF   (LG_OFF + 16 * LG_PITCH)  // 39040
#define ACC_OFF   (RED_OFF + 128)           // 39168
#define SMEM_FLOATS (ACC_OFF + 4)           // 39172 floats = 156688 B

__global__ __launch_bounds__(128)
void cpc_init(float* ws) {
    if (threadIdx.x < 2 * Sc) ws[threadIdx.x] = 0.0f;
}

__global__ __launch_bounds__(128)
void cpc_main(const float* __restrict__ base, const float* __restrict__ ctx,
              const int* __restrict__ seq_lens, const int* __restrict__ sample_ids,
              float* __restrict__ ws)
{
    extern __shared__ float smem[];
    float* negB = smem + NEG_OFF;
    float* aB   = smem + A_OFF;
    float* lg   = smem + LG_OFF;
    float* red  = smem + RED_OFF;
    float* acc2 = smem + ACC_OFF;

    const int tid = threadIdx.x;
    const int b  = blockIdx.z;          // batch
    const int i  = blockIdx.y + 1;      // step 1..S
    const int t0 = blockIdx.x * 16;     // t tile base
    const int seq = seq_lens[b];

    if (tid < 2) acc2[tid] = 0.0f;

    // ---- Stage neg_samples[b] into LDS via async global->LDS b128 (ASYNCcnt) ----
    // Thread tid owns gathered row k = tid. Layout: neg4[e4*128 + k] (float4).
    {
        const int k = tid;
        const long sid = (long)sample_ids[b * Kc + k];
        uint64_t ga = (uint64_t)(uintptr_t)(base + sid * (long)Ec);
        uint32_t la = (uint32_t)(uintptr_t)(negB) + (uint32_t)k * 16u;
        #pragma unroll 4
        for (int e4 = 0; e4 < Ec / 4; ++e4) {
            // copies 16 bytes per lane from the global address in ga to the LDS address in la; tracked on ASYNCcnt
            asm volatile("global_load_async_to_lds_b128 %0, %1, off"
                         :: "v"(la), "v"(ga) : "memory");
            ga += 16u;          // next float4 of this row
            la += (uint32_t)Kc * 16u;  // next e4 plane
        }
    }
    // ---- Stage A tile: ce[m][e] = ctx[b, t0+m, e, i-1] (stride-S reads; L2-resident) ----
    {
        const float* cb = ctx + ((long)b * Tc + t0) * (long)Ec * Sc + (i - 1);
        for (int idx = tid; idx < 16 * Ec; idx += 128) {
            int m = idx >> 8, e = idx & (Ec - 1);
            aB[m * A_PITCH + e] = cb[((long)m * Ec + e) * Sc];
        }
    }
    asm volatile("s_wait_asynccnt 0" ::: "memory");  // async LDS writes visible
    __syncthreads();

    // ---- pos[m] = dot(ce[m], base[b, t0+m+i]) : 8 threads per row ----
    {
        const int m = tid >> 3, j = tid & 7;
        float p = 0.0f;
        const int trow = t0 + m + i;
        if (trow < Tc) {
            const float* bp = base + ((long)b * Tc + trow) * Ec + j * 32;
            const float* ar = aB + m * A_PITCH + j * 32;
            #pragma unroll
            for (int e = 0; e < 32; ++e) p = fmaf(ar[e], bp[e], p);
        }
        red[m * 8 + j] = p;
    }
    __syncthreads();
    if ((tid & 7) == 0) {
        const int m = tid >> 3;
        float s = 0.0f;
        #pragma unroll
        for (int j = 0; j < 8; ++j) s += red[m * 8 + j];
        lg[m * LG_PITCH + 0] = s;
    }

    // ---- WMMA fp32 GEMM: neg logits (M=16, N=128, K=256) ----
    {
        const int lane = tid & 31;
        const int w = tid >> 5;                 // wave 0..3, owns N cols [w*32, w*32+32)
        const int mrow = lane & 15;             // A fragment: lanes hold M=0..15
        const int koff = (lane >> 4) << 1;      // lanes 16-31 hold K+2 (per ISA A layout)
        const float2* a2 = (const float2*)aB;
        const float2* n2 = (const float2*)negB;
        const int aRowHalf = mrow * (A_PITCH / 2);
        const int col0 = w * 32 + (lane & 15);

        v8f acc0 = {0.f,0.f,0.f,0.f,0.f,0.f,0.f,0.f};
        v8f acc1 = {0.f,0.f,0.f,0.f,0.f,0.f,0.f,0.f};

        #pragma unroll 4
        for (int e0 = 0; e0 < Ec; e0 += 4) {
            const int eh = (e0 + koff) >> 1;                 // float2 index along e
            // B element (e,k) lives at float4 slot [e>>2][k], float2 sub-slot (e&3)>>1
            const int bIdx = ((eh >> 1) * Kc + col0) * 2 + (eh & 1);
            float2 af = a2[aRowHalf + eh];   // ds_load_b64, conflict-free (pitch 260)
            float2 b0 = n2[bIdx];            // ds_load_b64, lanes stride 4 floats
            float2 b1 = n2[bIdx + 32];       // col0 + 16
            v2f a;   a[0]   = af.x; a[1]   = af.y;
            v2f vb0; vb0[0] = b0.x; vb0[1] = b0.y;
            v2f vb1; vb1[0] = b1.x; vb1[1] = b1.y;
            acc0 = __builtin_amdgcn_wmma_f32_16x16x4_f32(
                false, a, false, vb0, (short)0, acc0, false, false);
            acc1 = __builtin_amdgcn_wmma_f32_16x16x4_f32(
                false, a, false, vb1, (short)0, acc1, false, false);
        }

        // D layout: VGPR v, lanes 0-15 -> M=v, N=lane; lanes 16-31 -> M=v+8, N=lane-16
        const int mbase = (lane < 16) ? 0 : 8;
        const int nc = lane & 15;
        #pragma unroll
        for (int v = 0; v < 8; ++v) {
            lg[(v + mbase) * LG_PITCH + 1 + w * 32 + nc]      = acc0[v];
            lg[(v + mbase) * LG_PITCH + 1 + w * 32 + 16 + nc] = acc1[v];
        }
    }
    __syncthreads();

    // ---- per-row log-softmax over 129 logits; masked accumulate ----
    {
        const int m = tid >> 3, j = tid & 7;
        const float* row = lg + m * LG_PITCH;
        float mx = -3.402823466e38f;
        for (int c = j; c < Kc + 1; c += 8) mx = fmaxf(mx, row[c]);
        red[m * 8 + j] = mx;
        __syncthreads();
        float rmx = red[m * 8 + 0];
        #pragma unroll
        for (int j2 = 1; j2 < 8; ++j2) rmx = fmaxf(rmx, red[m * 8 + j2]);
        __syncthreads();
        float s = 0.0f;
        for (int c = j; c < Kc + 1; c += 8) s += expf(row[c] - rmx);
        red[m * 8 + j] = s;
        __syncthreads();
        if (j == 0) {
            float ss = 0.0f;
            #pragma unroll
            for (int j2 = 0; j2 < 8; ++j2) ss += red[m * 8 + j2];
            const float loss = rmx + logf(ss) - row[0];   // -log_softmax[...,0]
            const int t = t0 + m;
            const bool valid = (t + i < seq) && (t < Tc - i);
            atomicAdd(&acc2[0], valid ? loss : 0.0f);
            atomicAdd(&acc2[1], valid ? 1.0f : 0.0f);
        }
    }
    __syncthreads();
    if (tid == 0) {
        atomicAdd(&ws[i - 1],      acc2[0]);
        atomicAdd(&ws[Sc + i - 1], acc2[1]);
    }
}

__global__ __launch_bounds__(32)
void cpc_final(const float* __restrict__ ws, float* __restrict__ out) {
    if (threadIdx.x == 0) {
        float a = 0.0f;
        #pragma unroll
        for (int s = 0; s < Sc; ++s) a += ws[s] / ws[Sc + s];
        out[0] = a / (float)Sc;
    }
}

extern "C" void kernel_launch(void* const* d_in, const int* in_sizes, int n_in,
                              void* d_out, int out_size, void* d_ws, size_t ws_size,
                              hipStream_t stream) {
    const float* base = (const float*)d_in[0];   // (B,T,E) f32
    const float* ctx  = (const float*)d_in[1];   // (B,T,E,S) f32
    const int* seq    = (const int*)d_in[2];     // (B,) i32
    const int* sid    = (const int*)d_in[3];     // (B,K) i32
    float* out = (float*)d_out;
    float* wsf = (float*)d_ws;                   // 2*S floats of accumulators

    hipLaunchKernelGGL(cpc_init, dim3(1), dim3(128), 0, stream, wsf);

    dim3 grid(Tc / 16, Sc, Bc);                  // 32 x 5 x 64 = 10240 blocks
    size_t smem = (size_t)SMEM_FLOATS * sizeof(float);  // ~157 KB dynamic LDS
    hipLaunchKernelGGL(cpc_main, grid, dim3(128), smem, stream,
                       base, ctx, seq, sid, wsf);

    hipLaunchKernelGGL(cpc_final, dim3(1), dim3(32), 0, stream, wsf, out);
}